// GINEEncoder_27032524161222
// MI455X (gfx1250) — compile-verified
//
#include <hip/hip_runtime.h>

typedef __attribute__((ext_vector_type(2))) float v2f;
typedef __attribute__((ext_vector_type(4))) float v4f;
typedef __attribute__((ext_vector_type(8))) float v8f;

#define NF 128
#define GINE_BN_EPS 1e-5f

// ---------------- zero fill ----------------
__global__ void gine_zero_kernel(float* __restrict__ p, long long n) {
    long long i = (long long)blockIdx.x * blockDim.x + threadIdx.x;
    long long stride = (long long)gridDim.x * blockDim.x;
    for (; i < n; i += stride) p[i] = 0.0f;
}

// ---------------- scatter: agg[dst] += relu(x[src] + edge_attr) ----------------
// 32 threads per edge, 4 features per thread. edge_attr / edge_index are a pure
// 264 MB stream (> 192 MB L2): load non-temporally so the hot 25.6 MB x/agg
// node tables stay L2-resident.
__global__ void gine_scatter_kernel(const float* __restrict__ x,
                                    const float* __restrict__ ea,
                                    const long long* __restrict__ ei,
                                    float* __restrict__ agg, int E) {
    long long tid = (long long)blockIdx.x * blockDim.x + threadIdx.x;
    int e = (int)(tid >> 5);
    if (e >= E) return;
    int f = ((int)tid & 31) * 4;
    int s = (int)__builtin_nontemporal_load(ei + e);
    int d = (int)__builtin_nontemporal_load(ei + (long long)E + e);
    const v4f xv = *(const v4f*)(x + (long long)s * NF + f);
    const v4f av = __builtin_nontemporal_load((const v4f*)(ea + (long long)e * NF + f));
    float* dp = agg + (long long)d * NF + f;
    atomicAdd(dp + 0, fmaxf(xv.x + av.x, 0.0f));
    atomicAdd(dp + 1, fmaxf(xv.y + av.y, 0.0f));
    atomicAdd(dp + 2, fmaxf(xv.z + av.z, 0.0f));
    atomicAdd(dp + 3, fmaxf(xv.w + av.w, 0.0f));
}

// ---------------- WMMA fp32 GEMM: out = [relu]( (A1[+A2]) @ W + bias ) ----------------
// A: [M,128] row-major, W: [128,128] (k rows), out: [M,128].
// Block: 256 threads = 8 waves; each wave owns one 16-row M-tile, all 8 col tiles.
// W is staged into LDS K-pair interleaved: Wl[(k>>1)*256 + n*2 + (k&1)], so a
// B fragment (W[k0][n], W[k0+1][n]) is a single aligned ds_load_b64.
template <int ADD, int RELU>
__global__ void __launch_bounds__(256)
gine_gemm_kernel(const float* __restrict__ A1, const float* __restrict__ A2,
                 const float* __restrict__ W, const float* __restrict__ bias,
                 float* __restrict__ out, int M) {
    __shared__ float Wl[NF * NF];  // 64 KB

    // Cooperative interleaved stage: item = (pair p, column-pair ng).
    // LDS float4 at p*256 + 4*ng = {W[2p][2ng], W[2p+1][2ng], W[2p][2ng+1], W[2p+1][2ng+1]}
    for (int item = threadIdx.x; item < 64 * 64; item += blockDim.x) {
        const int p = item >> 6;          // k-pair 0..63
        const int ng = item & 63;         // column pair 0..63
        const v2f r0 = *(const v2f*)(W + (2 * p + 0) * NF + 2 * ng);
        const v2f r1 = *(const v2f*)(W + (2 * p + 1) * NF + 2 * ng);
        v4f q;
        q.x = r0.x; q.y = r1.x; q.z = r0.y; q.w = r1.y;
        *(v4f*)(Wl + p * 256 + 4 * ng) = q;
    }
    __syncthreads();

    const int wave = threadIdx.x >> 5;
    const int lane = threadIdx.x & 31;
    const int lm = lane & 15;    // M index within tile (A), N index (B/C)
    const int lh = lane >> 4;    // lane-half: selects K pair / +8 row half

    const int mtile = blockIdx.x * 8 + wave;
    const long long r0 = (long long)mtile * 16;
    if (r0 >= M) return;

    long long arow = r0 + lm;
    if (arow >= M) arow = M - 1;  // clamped read (stores are guarded)
    const float* a1p = A1 + arow * NF;
    const float* a2p = A2 + arow * NF;
    const float* wl = Wl + lh * 256 + lm * 2;  // this lane's base in interleaved W

    v8f acc[8] = {};

#pragma unroll 4
    for (int ks = 0; ks < 32; ++ks) {
        const int k0 = ks * 4 + 2 * lh;   // this lane-half's K pair
        v2f a = *(const v2f*)(a1p + k0);
        if (ADD) {
            const v2f a2 = *(const v2f*)(a2p + k0);
            a.x += a2.x;
            a.y += a2.y;
        }
        const float* wk = wl + ks * 512;  // pair index = 2*ks + lh
#pragma unroll
        for (int ct = 0; ct < 8; ++ct) {
            const v2f b = *(const v2f*)(wk + ct * 32);
            acc[ct] = __builtin_amdgcn_wmma_f32_16x16x4_f32(
                false, a, false, b, (short)0, acc[ct], false, false);
        }
    }

#pragma unroll
    for (int ct = 0; ct < 8; ++ct) {
        const int col = ct * 16 + lm;
        const float bv = bias[col];
#pragma unroll
        for (int i = 0; i < 8; ++i) {
            const long long row = r0 + i + 8 * lh;  // C layout: VGPR i, lane-half -> +8
            if (row < M) {
                float v = acc[ct][i] + bv;
                if (RELU) v = fmaxf(v, 0.0f);
                out[row * NF + col] = v;
            }
        }
    }
}

// ---------------- BN column statistics: sum / sumsq over rows ----------------
__global__ void gine_bn_stats_kernel(const float* __restrict__ h,
                                     float* __restrict__ gsum, float* __restrict__ gsq,
                                     int M) {
    __shared__ float s1[256];
    __shared__ float s2[256];
    const int c = threadIdx.x & 127;
    const int rh = threadIdx.x >> 7;
    float sum = 0.0f, sq = 0.0f;
    for (int r = blockIdx.x * 2 + rh; r < M; r += gridDim.x * 2) {
        float v = h[(long long)r * NF + c];
        sum += v;
        sq += v * v;
    }
    s1[threadIdx.x] = sum;
    s2[threadIdx.x] = sq;
    __syncthreads();
    if (threadIdx.x < 128) {
        atomicAdd(&gsum[c], s1[threadIdx.x] + s1[threadIdx.x + 128]);
        atomicAdd(&gsq[c], s2[threadIdx.x] + s2[threadIdx.x + 128]);
    }
}

__global__ void gine_bn_finalize_kernel(const float* __restrict__ gsum,
                                        const float* __restrict__ gsq,
                                        const float* __restrict__ gamma,
                                        const float* __restrict__ beta,
                                        float* __restrict__ scale,
                                        float* __restrict__ shift, int M) {
    const int c = threadIdx.x;
    const float invM = 1.0f / (float)M;
    const float mu = gsum[c] * invM;
    const float var = gsq[c] * invM - mu * mu;
    const float inv = rsqrtf(var + GINE_BN_EPS);
    const float sc = gamma[c] * inv;
    scale[c] = sc;
    shift[c] = beta[c] - mu * sc;
}

// o = relu(h * scale[c] + shift[c])
__global__ void gine_bn_apply_kernel(const float* __restrict__ h,
                                     const float* __restrict__ scale,
                                     const float* __restrict__ shift,
                                     float* __restrict__ o, long long n) {
    long long i = (long long)blockIdx.x * blockDim.x + threadIdx.x;
    long long stride = (long long)gridDim.x * blockDim.x;
    for (; i < n; i += stride) {
        int c = (int)(i & 127);
        o[i] = fmaxf(h[i] * scale[c] + shift[c], 0.0f);
    }
}

extern "C" void kernel_launch(void* const* d_in, const int* in_sizes, int n_in,
                              void* d_out, int out_size, void* d_ws, size_t ws_size,
                              hipStream_t stream) {
    const float* x        = (const float*)d_in[0];
    const long long* ei   = (const long long*)d_in[1];  // [2, E] int64
    const float* ea       = (const float*)d_in[2];
    const float* W1       = (const float*)d_in[3];
    const float* b1       = (const float*)d_in[4];
    const float* gamma    = (const float*)d_in[5];
    const float* beta     = (const float*)d_in[6];
    const float* W2       = (const float*)d_in[7];
    const float* b2       = (const float*)d_in[8];

    const int N = in_sizes[0] / NF;
    const int E = in_sizes[1] / 2;
    const int L = in_sizes[3] / (NF * NF);

    const long long nf = (long long)N * NF;
    float* agg   = (float*)d_ws;          // [N,128]  (also reused as BN'ed activations)
    float* h     = agg + nf;              // [N,128]
    float* y     = h + nf;                // [N,128]  intermediate layer output
    float* stats = y + nf;
    float* gsum  = stats;                 // [128]
    float* gsq   = stats + 128;           // [128]
    float* scale = stats + 256;           // [128]
    float* shift = stats + 384;           // [128]

    const int mtiles   = (N + 15) / 16;
    const int gblocks  = (mtiles + 7) / 8;
    const long long sthreads = (long long)E * 32;
    const int sblocks  = (int)((sthreads + 255) / 256);

    const float* xin = x;
    for (int l = 0; l < L; ++l) {
        float* lout = (l == L - 1) ? (float*)d_out : y;

        gine_zero_kernel<<<1024, 256, 0, stream>>>(agg, nf);
        gine_zero_kernel<<<1, 256, 0, stream>>>(gsum, 256);

        gine_scatter_kernel<<<sblocks, 256, 0, stream>>>(xin, ea, ei, agg, E);

        // h = (x + agg) @ W1 + b1
        gine_gemm_kernel<1, 0><<<gblocks, 256, 0, stream>>>(
            xin, agg, W1 + (long long)l * NF * NF, b1 + l * NF, h, N);

        gine_bn_stats_kernel<<<128, 256, 0, stream>>>(h, gsum, gsq, N);
        gine_bn_finalize_kernel<<<1, 128, 0, stream>>>(
            gsum, gsq, gamma + l * NF, beta + l * NF, scale, shift, N);

        // agg <- relu(BN(h))   (reuse agg buffer as GEMM2 input)
        gine_bn_apply_kernel<<<1024, 256, 0, stream>>>(h, scale, shift, agg, nf);

        // lout = relu(agg @ W2 + b2)   (outer relu is idempotent)
        gine_gemm_kernel<0, 1><<<gblocks, 256, 0, stream>>>(
            agg, agg, W2 + (long long)l * NF * NF, b2 + l * NF, lout, N);

        xin = lout;
    }
}